// H2GCN_68143951118647
// MI455X (gfx1250) — compile-verified
//
#include <hip/hip_runtime.h>

// ---------------------------------------------------------------------------
// H2GCN forward for MI455X (gfx1250, wave32).
//   N=100000 nodes, E=1600000 edges, D_IN=D_H=128, D_OUT=40.
// Dense GEMMs run on V_WMMA_F32_16X16X4_F32 (fp32 tensor path, exact
// precision vs the fp32 reference). Graph aggregation = fp32 global atomics
// on an L2-resident (51 MB) accumulator; that phase is the roofline limiter,
// not the GEMMs (~13 GFLOP total).
// LDS layouts are bank-conflict-free:
//   W quad-interleaved: Wlds[((k>>2)*NDIM + n)*4 + (k&3)]  -> b64 frag loads,
//     lanes 0-15 hit banks {4n,4n+1}, lanes 16-31 {4n+2,4n+3}.
//   A row stride padded to 132 floats -> lane m hits banks 4m(+2).
// ---------------------------------------------------------------------------

typedef __attribute__((ext_vector_type(2))) float v2f;
typedef __attribute__((ext_vector_type(8))) float v8f;

#define WAVES_PER_BLOCK 4
#define GEMM_BLOCK (WAVES_PER_BLOCK * 32)       // 128 threads
#define TILE_ROWS (WAVES_PER_BLOCK * 16)        // 64 output rows per block
#define DH 128
#define A_STRIDE 132                            // padded LDS row stride

// ---------------------------------------------------------------------------
// degree / normalization
// ---------------------------------------------------------------------------
__global__ void deg_init_kernel(int* __restrict__ cnt, int n) {
    int i = blockIdx.x * blockDim.x + threadIdx.x;
    if (i < n) cnt[i] = 1;                       // self-loop
}

__global__ void deg_accum_kernel(const long long* __restrict__ ei,
                                 int* __restrict__ cnt, int E) {
    int e = blockIdx.x * blockDim.x + threadIdx.x;
    if (e < E) atomicAdd(&cnt[(int)ei[(size_t)E + e]], 1);   // dst row
}

__global__ void dinv_kernel(const int* __restrict__ cnt,
                            float* __restrict__ dinv, int n) {
    int i = blockIdx.x * blockDim.x + threadIdx.x;
    if (i < n) dinv[i] = rsqrtf((float)cnt[i]);
}

// ---------------------------------------------------------------------------
// WMMA GEMM:  out[nrows x NCOLS] = concat(A0..A{NSEG-1}) * W[NSEG*128 x NCOLS]
//   Each A segment is a row-major [nrows x 128] block.
//   W staged once in LDS (quad-interleaved, zero-padded to NDIM columns).
//   A staged per 128-wide K segment (33 KB) -> small LDS, good occupancy.
//   Loop order ks-outer/nt-inner: one A frag feeds NTILES independent WMMAs.
//   WRITE_AGG: also emit agg[row][col] = out * dinv[row]^2 (GCN self-loop).
// ---------------------------------------------------------------------------
template <int NSEG, int NTILES, int NCOLS, bool BIAS, bool RELU, bool WRITE_AGG>
__global__ __launch_bounds__(GEMM_BLOCK)
void gemm_wmma_kernel(const float* __restrict__ A0,
                      const float* __restrict__ A1,
                      const float* __restrict__ A2,
                      const float* __restrict__ W,
                      const float* __restrict__ bias,
                      const float* __restrict__ dinv,
                      float* __restrict__ out,
                      float* __restrict__ agg,
                      int nrows) {
    constexpr int KDIM = NSEG * 128;
    constexpr int NDIM = NTILES * 16;
    extern __shared__ float smem[];
    float* Wlds = smem;                          // KDIM * NDIM floats (permuted)
    float* Alds = smem + KDIM * NDIM;            // WAVES_PER_BLOCK * 16 * A_STRIDE

    const int tid  = threadIdx.x;
    const int wave = tid >> 5;
    const int lane = tid & 31;

    // ---- stage all of W, quad-interleaved along k (zero-pad cols >= NCOLS) --
    for (int idx = tid; idx < KDIM * NDIM; idx += GEMM_BLOCK) {
        const int k = idx / NDIM, n = idx % NDIM;
        const float v = (n < NCOLS) ? W[(size_t)k * NCOLS + n] : 0.0f;
        Wlds[(size_t)(((k >> 2) * NDIM + n) << 2) + (k & 3)] = v;
    }

    const int row_base = blockIdx.x * TILE_ROWS + wave * 16;
    float* Aw = Alds + wave * (16 * A_STRIDE);

    const int n_lane = lane & 15;
    const int half   = lane >> 4;                // selects K-pair {0,1} vs {2,3}

    v8f c[NTILES] = {};                          // fp32 accumulators

    const float* segs[3] = {A0, A1, A2};
    for (int seg = 0; seg < NSEG; ++seg) {
        // ---- stage this wave's 16 rows of A segment (coalesced along k) ----
        const float* Aseg = segs[seg];
        for (int idx = lane; idx < 16 * 128; idx += 32) {
            const int m = idx >> 7, k = idx & 127;
            const int row = row_base + m;
            Aw[m * A_STRIDE + k] = (row < nrows) ? Aseg[(size_t)row * DH + k]
                                                 : 0.0f;
        }
        __syncthreads();   // seg 0: also covers Wlds staging

        const float* Wseg = Wlds + (size_t)seg * 128 * NDIM;
        #pragma unroll 4
        for (int ks = 0; ks < 32; ++ks) {
            const int k0 = ks * 4 + half * 2;
            // A frag: lane (half, m=n_lane) holds A[m][k0], A[m][k0+1]
            const v2f a = *(const v2f*)(Aw + n_lane * A_STRIDE + k0);
            #pragma unroll
            for (int nt = 0; nt < NTILES; ++nt) {
                // B frag: lane (half, n) holds W[k0][n], W[k0+1][n]  (b64, no
                // repacking: quad-interleaved layout keeps the pair adjacent)
                const v2f b = *(const v2f*)(
                    Wseg + (((size_t)ks * NDIM + nt * 16 + n_lane) << 2)
                         + half * 2);
                c[nt] = __builtin_amdgcn_wmma_f32_16x16x4_f32(
                            false, a, false, b, (short)0, c[nt], false, false);
            }
        }
        if (seg + 1 < NSEG) __syncthreads();     // before Alds overwrite
    }

    // ---- epilogue: C/D layout: VGPR r -> M = r + 8*half, N = lane&15 ----
    const bool full_rows = (row_base + 16 <= nrows);
    float di2[8];
    if (WRITE_AGG) {
        #pragma unroll
        for (int r = 0; r < 8; ++r) {
            const int row = row_base + r + 8 * half;
            const float di = (row < nrows) ? dinv[row] : 0.0f;
            di2[r] = di * di;
        }
    }
    #pragma unroll
    for (int nt = 0; nt < NTILES; ++nt) {
        const int col = nt * 16 + n_lane;
        const bool col_ok = (NDIM == NCOLS) || (col < NCOLS);  // compile-time
        if (!col_ok) continue;                   // only last partial tile
        const float bv = BIAS ? bias[col] : 0.0f;
        if (full_rows) {
            #pragma unroll
            for (int r = 0; r < 8; ++r) {
                const int row = row_base + r + 8 * half;
                float v = c[nt][r] + bv;
                if (RELU) v = fmaxf(v, 0.0f);
                out[(size_t)row * NCOLS + col] = v;
                if (WRITE_AGG) agg[(size_t)row * NCOLS + col] = v * di2[r];
            }
        } else {
            #pragma unroll
            for (int r = 0; r < 8; ++r) {
                const int row = row_base + r + 8 * half;
                if (row < nrows) {
                    float v = c[nt][r] + bv;
                    if (RELU) v = fmaxf(v, 0.0f);
                    out[(size_t)row * NCOLS + col] = v;
                    if (WRITE_AGG) agg[(size_t)row * NCOLS + col] = v * di2[r];
                }
            }
        }
    }
}

// ---------------------------------------------------------------------------
// edge scatter: agg[dst] += hw[src] * dinv[src]*dinv[dst]
//   32 lanes per edge -> full 512 B row per gather; 4 fp32 atomics per lane.
// ---------------------------------------------------------------------------
__device__ __forceinline__ void atomic_add_f32(float* p, float v) {
    __hip_atomic_fetch_add(p, v, __ATOMIC_RELAXED, __HIP_MEMORY_SCOPE_AGENT);
}

__global__ void scatter_edges_kernel(const long long* __restrict__ ei,
                                     const float* __restrict__ dinv,
                                     const float* __restrict__ hw,
                                     float* __restrict__ agg, int E) {
    long long t = (long long)blockIdx.x * blockDim.x + threadIdx.x;
    const int e    = (int)(t >> 5);
    const int lane = (int)(t & 31);
    if (e >= E) return;
    const int s = (int)ei[e];
    const int d = (int)ei[(size_t)E + e];
    const float norm = dinv[s] * dinv[d];
    const float4 v = *((const float4*)(hw + (size_t)s * DH) + lane);
    float* o = agg + (size_t)d * DH + lane * 4;
    atomic_add_f32(o + 0, v.x * norm);
    atomic_add_f32(o + 1, v.y * norm);
    atomic_add_f32(o + 2, v.z * norm);
    atomic_add_f32(o + 3, v.w * norm);
}

// ---------------------------------------------------------------------------
// in-place bias + relu over [N x 128]
// ---------------------------------------------------------------------------
__global__ void bias_relu_kernel(float* __restrict__ h,
                                 const float* __restrict__ b, int total) {
    int i = blockIdx.x * blockDim.x + threadIdx.x;
    if (i < total) h[i] = fmaxf(h[i] + b[i & (DH - 1)], 0.0f);
}

// ---------------------------------------------------------------------------
// launcher
// ---------------------------------------------------------------------------
extern "C" void kernel_launch(void* const* d_in, const int* in_sizes, int n_in,
                              void* d_out, int out_size, void* d_ws, size_t ws_size,
                              hipStream_t stream) {
    const float*     x       = (const float*)d_in[0];
    const long long* ei      = (const long long*)d_in[1];   // int64 [2,E]
    const float*     W_embed = (const float*)d_in[2];
    const float*     b_embed = (const float*)d_in[3];
    const float*     W_conv1 = (const float*)d_in[4];
    const float*     b_conv1 = (const float*)d_in[5];
    const float*     W_conv2 = (const float*)d_in[6];
    const float*     b_conv2 = (const float*)d_in[7];
    const float*     W_cls   = (const float*)d_in[8];
    const float*     b_cls   = (const float*)d_in[9];

    const int N = in_sizes[0] / DH;
    const int E = in_sizes[1] / 2;

    // workspace layout (16B-aligned slices)
    float* ws   = (float*)d_ws;
    int*   cnt  = (int*)ws;                       // N ints
    float* dinv = ws + N;                         // N floats
    float* h0   = ws + 2 * (size_t)N;             // N*128
    float* h1   = h0 + (size_t)N * DH;            // N*128 (also conv1 agg)
    float* h2   = h1 + (size_t)N * DH;            // N*128 (also conv2 agg)
    float* hw   = h2 + (size_t)N * DH;            // N*128 (GEMM scratch)

    const int gN  = (N + 255) / 256;
    const int gE  = (E + 255) / 256;
    const int gNd = ((int)((size_t)N * DH) + 255) / 256;
    const int gSc = (int)(((long long)E * 32 + 255) / 256);
    const int gGm = (N + TILE_ROWS - 1) / TILE_ROWS;
    const size_t aldsB   = (size_t)WAVES_PER_BLOCK * 16 * A_STRIDE * 4;      // 33 KB
    const size_t smem128 = (size_t)(128 * 128) * 4 + aldsB;                  // ~97 KB
    const size_t smem384 = (size_t)(384 * 48)  * 4 + aldsB;                  // ~107 KB

    // 1) degree -> dinv
    deg_init_kernel<<<gN, 256, 0, stream>>>(cnt, N);
    deg_accum_kernel<<<gE, 256, 0, stream>>>(ei, cnt, E);
    dinv_kernel<<<gN, 256, 0, stream>>>(cnt, dinv, N);

    // 2) h0 = relu(x @ W_embed + b_embed)
    gemm_wmma_kernel<1, 8, 128, true, true, false>
        <<<gGm, GEMM_BLOCK, smem128, stream>>>(
            x, x, x, W_embed, b_embed, nullptr, h0, nullptr, N);

    // 3) conv1: hw = h0 @ W_conv1 ; h1 seeded with self-loop term hw*dinv^2
    gemm_wmma_kernel<1, 8, 128, false, false, true>
        <<<gGm, GEMM_BLOCK, smem128, stream>>>(
            h0, h0, h0, W_conv1, nullptr, dinv, hw, h1, N);
    scatter_edges_kernel<<<gSc, 256, 0, stream>>>(ei, dinv, hw, h1, E);
    bias_relu_kernel<<<gNd, 256, 0, stream>>>(h1, b_conv1, (int)((size_t)N * DH));

    // 4) conv2
    gemm_wmma_kernel<1, 8, 128, false, false, true>
        <<<gGm, GEMM_BLOCK, smem128, stream>>>(
            h1, h1, h1, W_conv2, nullptr, dinv, hw, h2, N);
    scatter_edges_kernel<<<gSc, 256, 0, stream>>>(ei, dinv, hw, h2, E);
    bias_relu_kernel<<<gNd, 256, 0, stream>>>(h2, b_conv2, (int)((size_t)N * DH));

    // 5) out = concat(h0,h1,h2) @ W_cls + b_cls   (K=384, 40 cols padded to 48)
    gemm_wmma_kernel<3, 3, 40, true, false, false>
        <<<gGm, GEMM_BLOCK, smem384, stream>>>(
            h0, h1, h2, W_cls, b_cls, nullptr, (float*)d_out, nullptr, N);
}